// VOIDSpMM_70695161692195
// MI455X (gfx1250) — compile-verified
//
#include <hip/hip_runtime.h>
#include <hip/hip_bf16.h>

typedef __attribute__((ext_vector_type(2))) float v2f;
typedef __attribute__((ext_vector_type(8))) float v8f;

#define TILE_M 64
#define TILE_K 64
#define NCOLS  512
#define NCHUNK 128          // N columns per workgroup
#define A_STRIDE 68         // 64 + 4 pad: conflict-free b64 A-fragment loads
#define X2_STRIDE 288       // (128 cols * 2 K-interleave) + 32 pad:
                            // pair-row stride mod 64 == 32 -> hi/lo lane halves
                            // hit disjoint bank halves; conflict-free b64 loads

// ---------------------------------------------------------------------------
// Zero the output (harness poisons d_out with 0xAA; we accumulate atomically).
// ---------------------------------------------------------------------------
__global__ __launch_bounds__(256) void bsmm_zero_kernel(float4* __restrict__ y, int n4) {
    int i = blockIdx.x * blockDim.x + threadIdx.x;
    if (i < n4) y[i] = make_float4(0.f, 0.f, 0.f, 0.f);
}

// ---------------------------------------------------------------------------
// One workgroup = one sparse block  x  one 128-wide N chunk.
// 8 waves (wave32): wave w -> M-tile (w>>1), N-half (w&1) => 16x64 strip,
// i.e. 4 accumulators of 16x16 fed by v_wmma_f32_16x16x4_f32.
//
// x tile is staged K-pair-interleaved: Xs2[(K>>1)*X2_STRIDE + 2*col + (K&1)]
// so every B fragment (VGPR0=K even, VGPR1=K odd) is one aligned ds_load_b64.
// ---------------------------------------------------------------------------
__global__ __launch_bounds__(256) void bsmm_wmma_kernel(
    const float* __restrict__ values,      // [2048, 64, 64]
    const int*   __restrict__ block_rows,  // [2048]
    const int*   __restrict__ block_cols,  // [2048]
    const float* __restrict__ x,           // [8192, 512]
    float*       __restrict__ y)           // [8192, 512]
{
    __shared__ float As[TILE_M * A_STRIDE];          // 17.4 KB
    __shared__ float Xs2[(TILE_K / 2) * X2_STRIDE];  // 36.9 KB

    const int b   = blockIdx.y;
    const int n0  = blockIdx.x * NCHUNK;
    const int tid = threadIdx.x;

    const int br = block_rows[b];
    const int bc = block_cols[b];

    const float* Ablk = values + (size_t)b * TILE_M * TILE_K;
    const float* Xblk = x + (size_t)bc * TILE_K * NCOLS + n0;

    // ---- Stage A block into LDS (coalesced b128, padded rows) ----
    // 4096 floats / 256 threads = 4 float4 each.
    #pragma unroll
    for (int j = 0; j < 4; ++j) {
        int idx = j * 1024 + tid * 4;
        int r = idx >> 6;                      // /64
        int c = idx & 63;
        float4 v = *(const float4*)(Ablk + idx);
        *(float4*)(&As[r * A_STRIDE + c]) = v;
    }
    // ---- Stage x K-tile (64 x 128) into LDS, K-pair interleaved ----
    // 8192 floats / 256 threads = 8 float4 each (coalesced b128 global loads).
    #pragma unroll
    for (int j = 0; j < 8; ++j) {
        int idx = j * 1024 + tid * 4;
        int r = idx >> 7;                      // /128  (r = K index)
        int c = idx & 127;
        float4 v = *(const float4*)(Xblk + (size_t)r * NCOLS + c);
        float* dst = &Xs2[(r >> 1) * X2_STRIDE + 2 * c + (r & 1)];
        dst[0] = v.x; dst[2] = v.y; dst[4] = v.z; dst[6] = v.w;
    }
    __syncthreads();

    const int wave  = tid >> 5;                // 0..7 (wave32)
    const int lane  = tid & 31;
    const int l     = lane & 15;
    const int hi    = lane >> 4;               // lane half selects K pair / M half
    const int mtile = wave >> 1;               // 0..3
    const int nhalf = wave & 1;                // 0..1

    v8f acc0 = {}, acc1 = {}, acc2 = {}, acc3 = {};

    const float* Arow  = &As[(mtile * 16 + l) * A_STRIDE];
    const int    bcol2 = (nhalf * 64 + l) * 2;   // interleaved column offset

    #pragma unroll 4
    for (int k = 0; k < TILE_K; k += 4) {
        // A 16x4 fragment: VGPR j holds K = k + 2*hi + j  -> one b64 per lane
        v2f a = *(const v2f*)(Arow + k + 2 * hi);

        // B fragments: pair-row (k/2 + hi), nt tiles 32 dwords apart -> b64 each
        const float* xp = &Xs2[((k >> 1) + hi) * X2_STRIDE + bcol2];
        v2f b0 = *(const v2f*)(xp + 0);
        v2f b1 = *(const v2f*)(xp + 32);
        v2f b2 = *(const v2f*)(xp + 64);
        v2f b3 = *(const v2f*)(xp + 96);

        acc0 = __builtin_amdgcn_wmma_f32_16x16x4_f32(false, a, false, b0,
                                                     (short)0, acc0, false, false);
        acc1 = __builtin_amdgcn_wmma_f32_16x16x4_f32(false, a, false, b1,
                                                     (short)0, acc1, false, false);
        acc2 = __builtin_amdgcn_wmma_f32_16x16x4_f32(false, a, false, b2,
                                                     (short)0, acc2, false, false);
        acc3 = __builtin_amdgcn_wmma_f32_16x16x4_f32(false, a, false, b3,
                                                     (short)0, acc3, false, false);
    }

    // ---- Scatter-add into y: C/D layout M = vgpr + 8*hi, N = lane&15 ----
    const int orow0 = br * TILE_M + mtile * 16 + 8 * hi;
    const int ocol  = n0 + nhalf * 64 + l;
    #pragma unroll
    for (int i = 0; i < 8; ++i) {
        float* yr = y + (size_t)(orow0 + i) * NCOLS + ocol;
        atomicAdd(yr + 0,  acc0[i]);
        atomicAdd(yr + 16, acc1[i]);
        atomicAdd(yr + 32, acc2[i]);
        atomicAdd(yr + 48, acc3[i]);
    }
}

extern "C" void kernel_launch(void* const* d_in, const int* in_sizes, int n_in,
                              void* d_out, int out_size, void* d_ws, size_t ws_size,
                              hipStream_t stream) {
    const float* values     = (const float*)d_in[0];
    const int*   block_rows = (const int*)d_in[1];
    const int*   block_cols = (const int*)d_in[2];
    const float* x          = (const float*)d_in[3];
    float*       y          = (float*)d_out;

    const int n_blocks = in_sizes[1];          // 2048

    // 1) zero output
    int n4 = out_size / 4;                     // 8192*512/4 float4s
    bsmm_zero_kernel<<<(n4 + 255) / 256, 256, 0, stream>>>((float4*)y, n4);

    // 2) block-sparse WMMA matmul with atomic scatter-add
    dim3 grid(NCOLS / NCHUNK, n_blocks, 1);    // (4, 2048)
    bsmm_wmma_kernel<<<grid, 256, 0, stream>>>(values, block_rows, block_cols, x, y);
}